// ThreeInterp_70446053589571
// MI455X (gfx1250) — compile-verified
//
#include <hip/hip_runtime.h>

typedef __attribute__((ext_vector_type(2))) float v2f;
typedef __attribute__((ext_vector_type(8))) float v8f;

// monotone-sortable u32 of an f32 (ascending float -> ascending uint)
static __device__ __forceinline__ unsigned int sortable_u32(float f) {
    unsigned int b = __float_as_uint(f);
    return b ^ ((unsigned int)((int)b >> 31) | 0x80000000u);
}

// pack (sortable_key, index<2048) as an exact integer-valued double:
// value = s * 2048 + n  (43 bits, exact in f64; f64 order == u64 order)
static __device__ __forceinline__ double pack_key(float key, double nf) {
    return (double)sortable_u32(key) * 2048.0 + nf;
}

// branchless insert of x into ascending top-3 (k0 <= k1 <= k2):
// 3x v_min_num_f64 + 2x v_max_num_f64, no control flow, no cndmask chains
static __device__ __forceinline__ void ce3(double x, double &k0, double &k1, double &k2) {
    k2 = fmin(k2, fmax(k1, x));
    k1 = fmin(k1, fmax(k0, x));
    k0 = fmin(k0, x);
}

__global__ __launch_bounds__(32) void three_interp_wmma(
    const float* __restrict__ xyz1,
    const float* __restrict__ xyz2,
    const float* __restrict__ points1,
    const float* __restrict__ points2,
    float* __restrict__ out)
{
    constexpr int NQ = 8192, ND = 2048, CC = 256;
    constexpr int TILES = ND / 16;            // 128 db tiles of 16 points
    constexpr double KINF = 1.0e18;           // > any packed key (~8.8e12)

    __shared__ double sp[16][48];             // [row][lane16*3 + slot]
    __shared__ float  sw[16][3];
    __shared__ int    sx[16][3];

    const int lane = threadIdx.x;             // 0..31 (wave32)
    const int half = lane >> 4;
    const int l16  = lane & 15;
    const int b    = blockIdx.x >> 9;         // 512 query tiles per batch
    const int q0   = (blockIdx.x & 511) << 4;

    const float* x1 = xyz1 + (size_t)b * NQ * 3;
    const float* x2 = xyz2 + (size_t)b * ND * 3;

    // A fragment = -2 * query coords, 16x4 f32 WMMA layout:
    // lanes 0-15: (K=0,K=1) = (-2x_m, -2y_m); lanes 16-31: (K=2,K=3) = (-2z_m, 0)
    float qx = x1[(q0 + l16) * 3 + 0];
    float qy = x1[(q0 + l16) * 3 + 1];
    float qz = x1[(q0 + l16) * 3 + 2];
    v2f afrag;
    afrag.x = -2.0f * (half ? qz : qx);
    afrag.y = half ? 0.0f : (-2.0f * qy);

    // running per-lane top-3 packed keys for the 8 D-matrix rows this lane holds
    double k0[8], k1[8], k2[8];
#pragma unroll
    for (int r = 0; r < 8; ++r) { k0[r] = k1[r] = k2[r] = KINF; }

    for (int t = 0; t < TILES; ++t) {
        int n = t * 16 + l16;                 // this lane's db column
        float px = x2[n * 3 + 0];
        float py = x2[n * 3 + 1];
        float pz = x2[n * 3 + 2];
        float nrm = px * px + py * py + pz * pz;
        double nf = (double)n;
        // B fragment 4x16 f32: lanes 0-15 rows K=0,1 -> (x,y); lanes 16-31 rows K=2,3 -> (z,0)
        v2f bfrag;
        bfrag.x = half ? pz : px;
        bfrag.y = half ? 0.0f : py;
        v8f acc = {};                         // C = inline 0
        acc = __builtin_amdgcn_wmma_f32_16x16x4_f32(
            false, afrag, false, bfrag, (short)0, acc, false, false);
        // acc[r] = -2 q_row . p_col ; selection key = acc[r] + |p|^2 (|q|^2 is row-constant)
#pragma unroll
        for (int r = 0; r < 8; ++r) {
            double x = pack_key(acc[r] + nrm, nf);
            ce3(x, k0[r], k1[r], k2[r]);
        }
    }

    // dump per-lane candidates: D layout => row = half*8 + r, column group = l16
#pragma unroll
    for (int r = 0; r < 8; ++r) {
        int row = half * 8 + r;
        sp[row][l16 * 3 + 0] = k0[r];
        sp[row][l16 * 3 + 1] = k1[r];
        sp[row][l16 * 3 + 2] = k2[r];
    }
    __syncthreads();

    if (lane < 16) {
        int m = lane;
        double f0 = KINF, f1 = KINF, f2 = KINF;
        for (int j = 0; j < 48; ++j)
            ce3(sp[m][j], f0, f1, f2);
        // recover indices; recompute exact distances from coords
        int j0 = (int)((unsigned long long)f0 & 2047ull);
        int j1 = (int)((unsigned long long)f1 & 2047ull);
        int j2 = (int)((unsigned long long)f2 & 2047ull);
        float mx = x1[(q0 + m) * 3 + 0];
        float my = x1[(q0 + m) * 3 + 1];
        float mz = x1[(q0 + m) * 3 + 2];
        float qn = mx * mx + my * my + mz * mz;
        float e0, e1, e2;
        {
            float ax = x2[j0*3+0], ay = x2[j0*3+1], az = x2[j0*3+2];
            e0 = fmaxf(qn + (ax*ax + ay*ay + az*az) - 2.0f*(mx*ax + my*ay + mz*az), 1e-10f);
            float bx2 = x2[j1*3+0], by2 = x2[j1*3+1], bz2 = x2[j1*3+2];
            e1 = fmaxf(qn + (bx2*bx2 + by2*by2 + bz2*bz2) - 2.0f*(mx*bx2 + my*by2 + mz*bz2), 1e-10f);
            float cx = x2[j2*3+0], cy = x2[j2*3+1], cz = x2[j2*3+2];
            e2 = fmaxf(qn + (cx*cx + cy*cy + cz*cz) - 2.0f*(mx*cx + my*cy + mz*cz), 1e-10f);
        }
        float v0 = 1.0f / e0, v1 = 1.0f / e1, v2v = 1.0f / e2;
        float inv = 1.0f / (v0 + v1 + v2v);
        sw[m][0] = v0 * inv; sw[m][1] = v1 * inv; sw[m][2] = v2v * inv;
        sx[m][0] = j0; sx[m][1] = j1; sx[m][2] = j2;
    }
    __syncthreads();

    // interpolation + concat: 512 floats per query = 128 float4, 32 lanes x 2 iters
    const float* p2b = points2 + (size_t)b * ND * CC;
    for (int m = 0; m < 16; ++m) {
        int a0 = sx[m][0], a1 = sx[m][1], a2 = sx[m][2];
        float w0 = sw[m][0], w1 = sw[m][1], w2 = sw[m][2];
        const float4* r0  = (const float4*)(p2b + (size_t)a0 * CC);
        const float4* r1  = (const float4*)(p2b + (size_t)a1 * CC);
        const float4* r2  = (const float4*)(p2b + (size_t)a2 * CC);
        const float4* p1r = (const float4*)(points1 + ((size_t)b * NQ + q0 + m) * CC);
        float4* orow      = (float4*)(out + ((size_t)b * NQ + q0 + m) * (2 * CC));
#pragma unroll
        for (int j = 0; j < 2; ++j) {
            int f = j * 32 + lane;            // 0..63
            float4 g0 = r0[f], g1 = r1[f], g2 = r2[f];
            float4 o;
            o.x = w0 * g0.x + w1 * g1.x + w2 * g2.x;
            o.y = w0 * g0.y + w1 * g1.y + w2 * g2.y;
            o.z = w0 * g0.z + w1 * g1.z + w2 * g2.z;
            o.w = w0 * g0.w + w1 * g1.w + w2 * g2.w;
            orow[f] = o;                      // interpolated -> channels [0,256)
            orow[64 + f] = p1r[f];            // points1      -> channels [256,512)
        }
    }
}

extern "C" void kernel_launch(void* const* d_in, const int* in_sizes, int n_in,
                              void* d_out, int out_size, void* d_ws, size_t ws_size,
                              hipStream_t stream) {
    const float* xyz1    = (const float*)d_in[0];
    const float* xyz2    = (const float*)d_in[1];
    const float* points1 = (const float*)d_in[2];
    const float* points2 = (const float*)d_in[3];
    float* out = (float*)d_out;
    (void)in_sizes; (void)n_in; (void)out_size; (void)d_ws; (void)ws_size;

    dim3 grid(8 * 512);   // B * (N1/16) query tiles, one wave each
    dim3 block(32);
    hipLaunchKernelGGL(three_interp_wmma, grid, block, 0, stream,
                       xyz1, xyz2, points1, points2, out);
}